// EmbedBigraph_45732811767837
// MI455X (gfx1250) — compile-verified
//
#include <hip/hip_runtime.h>

typedef float v2f __attribute__((ext_vector_type(2)));
typedef float vf4 __attribute__((ext_vector_type(4)));
typedef float v8f __attribute__((ext_vector_type(8)));

#define N_  64
#define C_  32
#define M_  4000
#define L_  2000
#define D_  16
#define O_  32
#define K_  512       // C_*D_
#define KC  128       // K per staged chunk (8 channels x 16 d)
#define NCHUNK 4
#define GP  132       // padded LDS row stride (floats) for Gs
#define BP  132       // padded LDS row stride (floats) for Bs

#if defined(__AMDGCN__) && __has_builtin(__builtin_amdgcn_global_load_async_to_lds_b128)
#define USE_ASYNC_LDS 1
// Builtin expects pointers to 16-byte int vectors; AS1 = global (prints as
// __device__), AS3 = LDS.
typedef int vi4 __attribute__((vector_size(16)));
typedef __attribute__((address_space(1))) vi4* g_vi4_p;
typedef __attribute__((address_space(3))) vi4* l_vi4_p;
#else
#define USE_ASYNC_LDS 0
#endif

__global__ __launch_bounds__(256)
void embed_bigraph_wmma_f32(const float* __restrict__ x,
                            const float* __restrict__ W,
                            const float* __restrict__ bias,
                            const int*   __restrict__ idx,
                            float* __restrict__ out)
{
    __shared__ float Gs[N_ * GP];   // 64 x 128 gathered-x chunk (padded)  ~33.8 KB
    __shared__ float Bs[O_ * BP];   // 32 x 128 W[l] chunk (padded)        ~16.9 KB
    __shared__ int   sidx[D_];

    const int l    = blockIdx.x;
    const int t    = threadIdx.x;
    const int lane = t & 31;
    const int wave = t >> 5;

    if (t < D_) sidx[t] = idx[l * D_ + t];

    // wave -> 16x16 output tile: 4 n-tiles x 2 o-tiles
    const int ni = wave & 3;        // rows  ni*16 .. ni*16+15
    const int oi = wave >> 2;       // cols  oi*16 .. oi*16+15

    // ISA 16x4 f32 A layout: lanes 0-15 hold K0/K1, lanes 16-31 hold K2/K3
    const int row16 = lane & 15;
    const int koff  = (lane >> 4) << 1;   // 0 or 2

    v8f acc0 = {0.f, 0.f, 0.f, 0.f, 0.f, 0.f, 0.f, 0.f};
    v8f acc1 = {0.f, 0.f, 0.f, 0.f, 0.f, 0.f, 0.f, 0.f};

    const float* Wl = W + (size_t)l * (O_ * K_);

    for (int ch = 0; ch < NCHUNK; ++ch) {
        const int k0 = ch * KC;     // global K offset within [0,512)
        const int c0 = k0 >> 4;     // first channel of this chunk

        __syncthreads();            // previous chunk's LDS reads done (and sidx visible)

        // ---- stage W chunk: Bs[o][k] = W[l][o][k0+k]  (32 x 128 floats)
#if USE_ASYNC_LDS
        // CDNA5 async copy: global -> LDS on the ASYNCcnt path, overlaps the
        // x-gather below (which runs on LOADcnt). 16B per lane per issue.
        #pragma unroll
        for (int i = 0; i < 4; ++i) {
            int j = t + i * 256;            // float4 id 0..1023
            int o = j >> 5;                 // 32 float4 per row
            int k = (j & 31) << 2;
            __builtin_amdgcn_global_load_async_to_lds_b128(
                (g_vi4_p)(Wl + o * K_ + k0 + k),
                (l_vi4_p)(&Bs[o * BP + k]),
                /*offset=*/0, /*cpol=*/0);
        }
#else
        #pragma unroll
        for (int i = 0; i < 4; ++i) {
            int j = t + i * 256;
            int o = j >> 5;
            int k = (j & 31) << 2;
            vf4 v = __builtin_nontemporal_load((const vf4*)(Wl + o * K_ + k0 + k));
            *(vf4*)(&Bs[o * BP + k]) = v;   // 16B aligned (k%4==0, BP%4==0)
        }
#endif

        // ---- stage gathered x chunk: Gs[n][k] = x[n][c0 + k/16][ sidx[k%16] ]
        #pragma unroll
        for (int i = 0; i < 32; ++i) {
            int e = t + i * 256;            // 0..8191
            int n = e >> 7;
            int k = e & 127;
            int c = c0 + (k >> 4);
            int d = k & 15;
            Gs[n * GP + k] = x[(size_t)n * (C_ * M_) + c * M_ + sidx[d]];
        }

#if USE_ASYNC_LDS
        asm volatile("s_wait_asynccnt 0x0" ::: "memory");  // our W copies landed in LDS
#endif
        __syncthreads();

        // ---- 32 chained V_WMMA_F32_16X16X4_F32 steps, 2 accumulators for ILP
        const float* ga = &Gs[(ni * 16 + row16) * GP + koff];
        const float* gb = &Bs[(oi * 16 + row16) * BP + koff];
        #pragma unroll 4
        for (int kk = 0; kk < KC; kk += 8) {
            v2f a0 = *(const v2f*)(ga + kk);       // 8B aligned
            v2f b0 = *(const v2f*)(gb + kk);
            acc0 = __builtin_amdgcn_wmma_f32_16x16x4_f32(
                       false, a0, false, b0, (short)0, acc0, false, false);
            v2f a1 = *(const v2f*)(ga + kk + 4);
            v2f b1 = *(const v2f*)(gb + kk + 4);
            acc1 = __builtin_amdgcn_wmma_f32_16x16x4_f32(
                       false, a1, false, b1, (short)0, acc1, false, false);
        }
    }

    // ---- epilogue: merge accumulators, add bias, store.
    // C/D layout: N = lane%16 (all 8 VGPRs), M = vgpr_j + 8*(lane>=16).
    v8f acc = acc0 + acc1;
    const int o  = oi * 16 + row16;
    const float bo = bias[l * O_ + o];
    const int nbase = ni * 16 + ((lane >> 4) << 3);
    #pragma unroll
    for (int j = 0; j < 8; ++j) {
        int n = nbase + j;
        out[(size_t)n * (O_ * L_) + (size_t)o * L_ + l] = acc[j] + bo;
    }
}

extern "C" void kernel_launch(void* const* d_in, const int* in_sizes, int n_in,
                              void* d_out, int out_size, void* d_ws, size_t ws_size,
                              hipStream_t stream) {
    (void)in_sizes; (void)n_in; (void)out_size; (void)d_ws; (void)ws_size;
    const float* x    = (const float*)d_in[0];
    const float* W    = (const float*)d_in[1];
    const float* bias = (const float*)d_in[2];
    const int*   idx  = (const int*)d_in[3];
    float* out = (float*)d_out;

    embed_bigraph_wmma_f32<<<L_, 256, 0, stream>>>(x, W, bias, idx, out);
}